// GraphSAGE_32727650795728
// MI455X (gfx1250) — compile-verified
//
#include <hip/hip_runtime.h>

typedef __attribute__((ext_vector_type(2))) float v2f;
typedef __attribute__((ext_vector_type(8))) float v8f;

// Packed weight layout: element (k, n) of the stacked weight [KTOT x 64]
// is stored at ((k>>1)*64 + n)*2 + (k&1), so a B fragment (k, k+1 at fixed n)
// is one aligned 8-byte load.
__device__ __forceinline__ size_t wpack(int k, int n) {
  return ((size_t)(k >> 1) * 64 + n) * 2 + (k & 1);
}

// ---------------------------------------------------------------- utilities

__global__ void zero_f32(float* __restrict__ p, long n) {
  long i = (long)blockIdx.x * blockDim.x + threadIdx.x;
  if (i < n) p[i] = 0.0f;
}

// h0[v][0:16] = [x(11) | pos(3) | 0 | 0]
__global__ void build_h0(const float* __restrict__ x, const float* __restrict__ pos,
                         float* __restrict__ h0, int n) {
  int t = blockIdx.x * blockDim.x + threadIdx.x;
  if (t >= n * 16) return;
  int v = t >> 4, c = t & 15;
  float val = 0.0f;
  if (c < 11)      val = x[v * 11 + c];
  else if (c < 14) val = pos[v * 3 + (c - 11)];
  h0[t] = val;
}

// Wc1 packed [32 x 64]: rows 0..13 = W1_l, 14..15 = 0, 16..29 = W1_r, 30..31 = 0
__global__ void build_Wc1(const float* __restrict__ Wl, const float* __restrict__ Wr,
                          float* __restrict__ Wc) {
  int t = blockIdx.x * blockDim.x + threadIdx.x;
  if (t >= 32 * 64) return;
  int r = t >> 6, c = t & 63;
  float v = 0.0f;
  if (r < 14)                 v = Wl[r * 64 + c];
  else if (r >= 16 && r < 30) v = Wr[(r - 16) * 64 + c];
  Wc[wpack(r, c)] = v;
}

// Wc2 packed [128 x 64] = [W2_l ; W2_r]
__global__ void build_Wc2(const float* __restrict__ Wl, const float* __restrict__ Wr,
                          float* __restrict__ Wc) {
  int t = blockIdx.x * blockDim.x + threadIdx.x;
  if (t >= 128 * 64) return;
  int r = t >> 6, c = t & 63;
  Wc[wpack(r, c)] = (r < 64) ? Wl[r * 64 + c] : Wr[(r - 64) * 64 + c];
}

__global__ void edge_degree(const int* __restrict__ dst, float* __restrict__ deg, int e) {
  int t = blockIdx.x * blockDim.x + threadIdx.x;
  if (t < e) atomicAdd(&deg[dst[t]], 1.0f);
}

__global__ void node_count(const int* __restrict__ batch, float* __restrict__ cnt, int n) {
  int t = blockIdx.x * blockDim.x + threadIdx.x;
  if (t < n) atomicAdd(&cnt[batch[t]], 1.0f);
}

// -------------------------------------------------------- edge scatter-adds

// D = 16 features, 4 threads per edge, float4 gather + 4 f32 atomics
__global__ void scatter16(const float* __restrict__ h, const int* __restrict__ src,
                          const int* __restrict__ dst, float* __restrict__ agg, int e) {
  long t = (long)blockIdx.x * blockDim.x + threadIdx.x;
  if (t >= (long)e * 4) return;
  int ei = (int)(t >> 2);
  int c  = (int)(t & 3) << 2;
  int s = src[ei], d = dst[ei];
  const float4 m = *reinterpret_cast<const float4*>(h + (size_t)s * 16 + c);
  float* a = agg + (size_t)d * 16 + c;
  atomicAdd(a + 0, m.x); atomicAdd(a + 1, m.y);
  atomicAdd(a + 2, m.z); atomicAdd(a + 3, m.w);
}

// D = 64 features, 16 threads per edge
__global__ void scatter64(const float* __restrict__ h, const int* __restrict__ src,
                          const int* __restrict__ dst, float* __restrict__ agg, int e) {
  long t = (long)blockIdx.x * blockDim.x + threadIdx.x;
  if (t >= (long)e * 16) return;
  int ei = (int)(t >> 4);
  int c  = (int)(t & 15) << 2;
  int s = src[ei], d = dst[ei];
  const float4 m = *reinterpret_cast<const float4*>(h + (size_t)s * 64 + c);
  float* a = agg + (size_t)d * 64 + c;
  atomicAdd(a + 0, m.x); atomicAdd(a + 1, m.y);
  atomicAdd(a + 2, m.z); atomicAdd(a + 3, m.w);
}

// ------------------------------------------------- fused SAGE update (WMMA)
// out[v] = relu( [agg[v]/max(deg,1) | hin[v]] @ Wc + bias )
// One wave per 16-node tile; 4 f32 accumulators cover the 64 output cols.
// A: f32 16x4 fragments from global (aligned float2, 1/deg folded in).
// B: pre-swizzled stacked weights async-staged into LDS; each fragment is
//    one ds_load_b64.
// Layer 1 writes h1; layer 2 (pooled != nullptr) scatters into pooled sums.
template <int KHALF>
__global__ __launch_bounds__(256) void sage_update(
    const float* __restrict__ agg, const float* __restrict__ hin,
    const float* __restrict__ deg, const float* __restrict__ Wc,
    const float* __restrict__ bias, float* __restrict__ hout,
    float* __restrict__ pooled, const int* __restrict__ batch, int nnodes) {
  constexpr int KTOT = 2 * KHALF;
  constexpr int WBYTES = KTOT * 64 * 4;  // 8 KB (layer1) / 32 KB (layer2)
  __shared__ float sW[KTOT * 64];
  const int tid = threadIdx.x;

  // CDNA5 async copy global -> LDS (ASYNCcnt-tracked, 16 B per lane).
  {
    const unsigned ldsbase = (unsigned)(uintptr_t)(void*)sW;
    const char* gsrc = (const char*)Wc;
#pragma unroll
    for (int i = tid * 16; i < WBYTES; i += 256 * 16) {
      unsigned laddr = ldsbase + i;
      const char* gaddr = gsrc + i;
      asm volatile("global_load_async_to_lds_b128 %0, %1, off"
                   :: "v"(laddr), "v"(gaddr)
                   : "memory");
    }
    asm volatile("s_wait_asynccnt 0x0" ::: "memory");
  }
  __syncthreads();

  const int lane = tid & 31;
  const int wave = tid >> 5;
  const int rowbase = blockIdx.x * 128 + wave * 16;
  const int r0 = rowbase + (lane & 15);
  const int rr = (r0 < nnodes) ? r0 : (nnodes - 1);  // clamp: keeps EXEC all-1s
  const float dv = deg[rr];
  const float inv = 1.0f / (dv > 1.0f ? dv : 1.0f);
  const int khi = (lane >> 4) << 1;  // 0 (lanes 0-15) or 2 (lanes 16-31)
  const int nc = lane & 15;

  v8f acc0 = {}, acc1 = {}, acc2 = {}, acc3 = {};

  for (int k0 = 0; k0 < KTOT; k0 += 4) {
    const bool lhs = (k0 < KHALF);
    const float* abase = lhs ? agg : hin;
    const int kloc = lhs ? k0 : (k0 - KHALF);
    const float scale = lhs ? inv : 1.0f;

    const float2 av =
        *reinterpret_cast<const float2*>(abase + (size_t)rr * KHALF + kloc + khi);
    v2f a;
    a.x = av.x * scale;
    a.y = av.y * scale;

    // Packed layout: (kb, kb+1) at fixed n is one aligned 8-byte LDS load.
    const int kb = k0 + khi;
    const float* brow = sW + (size_t)(kb >> 1) * 128;  // *64 cols * 2 interleave
    const v2f b0 = *reinterpret_cast<const v2f*>(brow + (size_t)(nc) * 2);
    const v2f b1 = *reinterpret_cast<const v2f*>(brow + (size_t)(16 + nc) * 2);
    const v2f b2 = *reinterpret_cast<const v2f*>(brow + (size_t)(32 + nc) * 2);
    const v2f b3 = *reinterpret_cast<const v2f*>(brow + (size_t)(48 + nc) * 2);

    acc0 = __builtin_amdgcn_wmma_f32_16x16x4_f32(false, a, false, b0, (short)0, acc0, false, false);
    acc1 = __builtin_amdgcn_wmma_f32_16x16x4_f32(false, a, false, b1, (short)0, acc1, false, false);
    acc2 = __builtin_amdgcn_wmma_f32_16x16x4_f32(false, a, false, b2, (short)0, acc2, false, false);
    acc3 = __builtin_amdgcn_wmma_f32_16x16x4_f32(false, a, false, b3, (short)0, acc3, false, false);
  }

  // C/D layout: VGPR r -> M = r (+8 for lanes 16-31), N = nc (+16 per tile)
  const int mhi = (lane >> 4) << 3;
#pragma unroll
  for (int r = 0; r < 8; ++r) {
    const int row = rowbase + mhi + r;
    if (row >= nnodes) continue;
    float v0 = acc0[r] + bias[nc];
    float v1 = acc1[r] + bias[16 + nc];
    float v2 = acc2[r] + bias[32 + nc];
    float v3 = acc3[r] + bias[48 + nc];
    v0 = v0 > 0.0f ? v0 : 0.0f;
    v1 = v1 > 0.0f ? v1 : 0.0f;
    v2 = v2 > 0.0f ? v2 : 0.0f;
    v3 = v3 > 0.0f ? v3 : 0.0f;
    if (pooled) {
      const int g = batch[row];
      float* p = pooled + (size_t)g * 64;
      atomicAdd(p + nc, v0);
      atomicAdd(p + 16 + nc, v1);
      atomicAdd(p + 32 + nc, v2);
      atomicAdd(p + 48 + nc, v3);
    } else {
      float* o = hout + (size_t)row * 64;
      o[nc] = v0; o[16 + nc] = v1; o[32 + nc] = v2; o[48 + nc] = v3;
    }
  }
}

// ------------------------------------------------------------ final readout
// out[g] = (pooled[g] . W_lin) / max(cnt,1) + b_lin
__global__ void readout(const float* __restrict__ pooled, const float* __restrict__ cnt,
                        const float* __restrict__ Wlin, const float* __restrict__ blin,
                        float* __restrict__ out, int g) {
  int t = blockIdx.x * blockDim.x + threadIdx.x;
  if (t >= g) return;
  float s = 0.0f;
#pragma unroll
  for (int k = 0; k < 64; ++k) s += pooled[(size_t)t * 64 + k] * Wlin[k];
  float c = cnt[t];
  c = c > 1.0f ? c : 1.0f;
  out[t] = s / c + blin[0];
}

// ------------------------------------------------------------------- launch

extern "C" void kernel_launch(void* const* d_in, const int* in_sizes, int n_in,
                              void* d_out, int out_size, void* d_ws, size_t ws_size,
                              hipStream_t stream) {
  const float* x    = (const float*)d_in[0];
  const float* pos  = (const float*)d_in[1];
  const int* eidx   = (const int*)d_in[2];
  const int* batch  = (const int*)d_in[3];
  const float* W1l  = (const float*)d_in[4];
  const float* W1r  = (const float*)d_in[5];
  const float* b1   = (const float*)d_in[6];
  const float* W2l  = (const float*)d_in[7];
  const float* W2r  = (const float*)d_in[8];
  const float* b2   = (const float*)d_in[9];
  const float* Wlin = (const float*)d_in[10];
  const float* blin = (const float*)d_in[11];
  float* out = (float*)d_out;

  const int N = in_sizes[0] / 11;   // 100000
  const int E = in_sizes[2] / 2;    // 3200000
  const int G = out_size;           // 5000
  const int* src = eidx;
  const int* dst = eidx + E;

  // workspace layout (floats)
  float* ws = (float*)d_ws;
  size_t o = 0;
  float* h0  = ws + o; o += (size_t)N * 16;
  float* h1  = ws + o; o += (size_t)N * 64;
  float* Wc1 = ws + o; o += 32 * 64;
  float* Wc2 = ws + o; o += 128 * 64;
  float* zbeg = ws + o;  // everything below must start zeroed
  float* deg    = ws + o; o += (size_t)N;
  float* agg1   = ws + o; o += (size_t)N * 16;
  float* agg2   = ws + o; o += (size_t)N * 64;
  float* pooled = ws + o; o += (size_t)G * 64;
  float* cnt    = ws + o; o += (size_t)G;
  const long zlen = (long)(ws + o - zbeg);

  const int T = 256;
  zero_f32<<<(int)((zlen + T - 1) / T), T, 0, stream>>>(zbeg, zlen);
  build_h0<<<(N * 16 + T - 1) / T, T, 0, stream>>>(x, pos, h0, N);
  build_Wc1<<<(32 * 64 + T - 1) / T, T, 0, stream>>>(W1l, W1r, Wc1);
  build_Wc2<<<(128 * 64 + T - 1) / T, T, 0, stream>>>(W2l, W2r, Wc2);
  edge_degree<<<(E + T - 1) / T, T, 0, stream>>>(dst, deg, E);
  node_count<<<(N + T - 1) / T, T, 0, stream>>>(batch, cnt, N);

  // layer 1
  scatter16<<<(int)(((long)E * 4 + T - 1) / T), T, 0, stream>>>(h0, src, dst, agg1, E);
  sage_update<16><<<(N + 127) / 128, T, 0, stream>>>(agg1, h0, deg, Wc1, b1, h1,
                                                     nullptr, nullptr, N);
  // layer 2 (epilogue fused into pooled scatter)
  scatter64<<<(int)(((long)E * 16 + T - 1) / T), T, 0, stream>>>(h1, src, dst, agg2, E);
  sage_update<64><<<(N + 127) / 128, T, 0, stream>>>(agg2, h1, deg, Wc2, b2, nullptr,
                                                     pooled, batch, N);
  // readout
  readout<<<(G + T - 1) / T, T, 0, stream>>>(pooled, cnt, Wlin, blin, out, G);
}